// nf_encode_3676492005549
// MI455X (gfx1250) — compile-verified
//
#include <hip/hip_runtime.h>
#include <hip/hip_bf16.h>

// ---------------- problem constants ----------------
#define TT   128
#define NN   1500
#define HH   64
#define KS   10
#define C1   8
#define C2   16
#define L1   119          // T - (KS-1)
#define L2   110          // L1 - (KS-1)
#define FCD  1760         // C2*L2
#define BB   16
#define CNT1 178500.0f    // NN*L1
#define CNT2 165000.0f    // NN*L2
#define CNT3 1500.0f
#define BN_EPS 1e-5f

// ---------------- workspace layout (float offsets) ----------------
#define S1SUM  0
#define S1SQ   8
#define S1S    16
#define S1T    24
#define S2SUM  32
#define S2SQ   48
#define S2S    64
#define S2T    80
#define S3SUM  96
#define S3SQ   160
#define S3S    224
#define S3T    288
#define ROWSUM 352                  // 1500 floats
#define STATS_FLOATS 2048
#define H1OFF  2048                 // 8*119*1500 = 1,428,000
#define H2OFF  (H1OFF + C1*L1*NN)   // 1,430,048 ; 1500*1760 = 2,640,000
#define X3OFF  (H2OFF + NN*FCD)     // 4,070,048 ; 96,000
#define AAOFF  (X3OFF + NN*HH)      // 4,166,048 ; 96,000
#define RBOFF  (AAOFF + NN*HH)      // 4,262,048 ; 96,000
#define SUPOFF 2048                 // reuses h1/h2 region; 2,250,000 floats

typedef __attribute__((ext_vector_type(2))) float v2f;
typedef __attribute__((ext_vector_type(8))) float v8f;

// ------------------------------------------------------------------
// K1: conv1 + relu -> h1[c][l][n], accumulate BN1 stats
// ------------------------------------------------------------------
__global__ __launch_bounds__(256)
void k_conv1(const float* __restrict__ x, const float* __restrict__ w1,
             const float* __restrict__ b1, float* __restrict__ ws) {
  __shared__ float wsh[C1 * KS];
  __shared__ float ssum[C1], ssq[C1];
  int t = threadIdx.x;
  if (t < C1 * KS) wsh[t] = w1[t];
  if (t < C1) { ssum[t] = 0.f; ssq[t] = 0.f; }
  __syncthreads();

  int idx = blockIdx.x * 256 + t;
  bool act = idx < NN * L1;
  float v[C1];
  if (act) {
    int n = idx % NN;
    int l = idx / NN;
    float acc[C1];
#pragma unroll
    for (int c = 0; c < C1; c++) acc[c] = b1[c];
#pragma unroll
    for (int k = 0; k < KS; k++) {
      float xv = x[(l + k) * NN + n];
#pragma unroll
      for (int c = 0; c < C1; c++) acc[c] = fmaf(wsh[c * KS + k], xv, acc[c]);
    }
    float* h1 = ws + H1OFF;
#pragma unroll
    for (int c = 0; c < C1; c++) {
      float r = fmaxf(acc[c], 0.f);
      v[c] = r;
      h1[(c * L1 + l) * NN + n] = r;
    }
#pragma unroll
    for (int c = 0; c < C1; c++) {
      atomicAdd(&ssum[c], v[c]);
      atomicAdd(&ssq[c], v[c] * v[c]);
    }
  }
  __syncthreads();
  if (t < C1) {
    atomicAdd(&ws[S1SUM + t], ssum[t]);
    atomicAdd(&ws[S1SQ + t], ssq[t]);
  }
}

// ------------------------------------------------------------------
// K2: finalize BN1 -> scale/shift
// ------------------------------------------------------------------
__global__ void k_fin1(const float* __restrict__ g, const float* __restrict__ b,
                       float* __restrict__ ws) {
  int c = threadIdx.x;
  if (c < C1) {
    float m   = ws[S1SUM + c] / CNT1;
    float var = ws[S1SQ + c] / CNT1 - m * m;
    float s   = g[c] * rsqrtf(var + BN_EPS);
    ws[S1S + c] = s;
    ws[S1T + c] = b[c] - m * s;
  }
}

// ------------------------------------------------------------------
// K3: conv2(bn1(h1)) + relu -> h2[n][c2*110+l2], accumulate BN2 stats
// ------------------------------------------------------------------
__global__ __launch_bounds__(256)
void k_conv2(const float* __restrict__ w2, const float* __restrict__ b2,
             float* __restrict__ ws) {
  __shared__ float wsh[C2 * C1 * KS];   // 1280
  __shared__ float s1[C1], t1[C1], ssum[C2], ssq[C2];
  int t = threadIdx.x;
  for (int i = t; i < C2 * C1 * KS; i += 256) wsh[i] = w2[i];
  if (t < C1) { s1[t] = ws[S1S + t]; t1[t] = ws[S1T + t]; }
  if (t < C2) { ssum[t] = 0.f; ssq[t] = 0.f; }
  __syncthreads();

  int idx = blockIdx.x * 256 + t;
  bool act = idx < NN * L2;
  if (act) {
    int n = idx % NN;
    int l2 = idx / NN;
    float acc[C2];
#pragma unroll
    for (int c = 0; c < C2; c++) acc[c] = b2[c];
    const float* h1 = ws + H1OFF;
    for (int c1 = 0; c1 < C1; c1++) {
      float s = s1[c1], tt = t1[c1];
#pragma unroll
      for (int k = 0; k < KS; k++) {
        float xv = fmaf(s, h1[(c1 * L1 + l2 + k) * NN + n], tt);
#pragma unroll
        for (int c2 = 0; c2 < C2; c2++)
          acc[c2] = fmaf(wsh[(c2 * C1 + c1) * KS + k], xv, acc[c2]);
      }
    }
    float* h2 = ws + H2OFF;
#pragma unroll
    for (int c2 = 0; c2 < C2; c2++) {
      float r = fmaxf(acc[c2], 0.f);
      h2[n * FCD + c2 * L2 + l2] = r;
      atomicAdd(&ssum[c2], r);
      atomicAdd(&ssq[c2], r * r);
    }
  }
  __syncthreads();
  if (t < C2) {
    atomicAdd(&ws[S2SUM + t], ssum[t]);
    atomicAdd(&ws[S2SQ + t], ssq[t]);
  }
}

// ------------------------------------------------------------------
// K4: finalize BN2
// ------------------------------------------------------------------
__global__ void k_fin2(const float* __restrict__ g, const float* __restrict__ b,
                       float* __restrict__ ws) {
  int c = threadIdx.x;
  if (c < C2) {
    float m   = ws[S2SUM + c] / CNT2;
    float var = ws[S2SQ + c] / CNT2 - m * m;
    float s   = g[c] * rsqrtf(var + BN_EPS);
    ws[S2S + c] = s;
    ws[S2T + c] = b[c] - m * s;
  }
}

// ------------------------------------------------------------------
// K4b: apply BN2 to h2 in place
// ------------------------------------------------------------------
__global__ __launch_bounds__(256)
void k_bn2(float* __restrict__ ws) {
  int idx = blockIdx.x * 256 + threadIdx.x;
  if (idx >= NN * FCD) return;
  int k = idx % FCD;
  int c = k / L2;
  float* h2 = ws + H2OFF;
  h2[idx] = fmaf(ws[S2S + c], h2[idx], ws[S2T + c]);
}

// ------------------------------------------------------------------
// K5: FC GEMM via V_WMMA_F32_16X16X4_F32, one wave per 16x16 tile.
//     x3raw = relu(x2 @ fc_w^T + fc_b); accumulate BN3 stats.
// A 16x4 layout: lanes 0-15 hold K={0,1}, lanes 16-31 hold K={2,3}.
// C/D layout: VGPR r -> row r + 8*(lane>=16), col lane&15.
// Out-of-range A rows are CLAMPED (not zeroed): their D rows are garbage
// but the epilogue never stores/accumulates them. This keeps the hot loop
// divergence-free (no exec-mask save/restore) and lets loads pipeline
// ahead of the WMMA via explicit double-buffering.
// ------------------------------------------------------------------
__global__ __launch_bounds__(32)
void k_fc_wmma(const float* __restrict__ fcw, const float* __restrict__ fcb,
               float* __restrict__ ws) {
  int lane = threadIdx.x;
  int n0 = blockIdx.x * 16;
  int h0 = blockIdx.y * 16;
  int lr = lane & 15;
  int hi = lane >> 4;
  int koff = hi * 2;

  const float* x2 = ws + H2OFF;
  int arow = n0 + lr;
  int arow_c = (arow < NN) ? arow : (NN - 1);   // clamp; garbage rows unused
  const v2f* ap = (const v2f*)(x2 + (size_t)arow_c * FCD + koff);
  const v2f* bp = (const v2f*)(fcw + (size_t)(h0 + lr) * FCD + koff);

  v8f acc = {0.f, 0.f, 0.f, 0.f, 0.f, 0.f, 0.f, 0.f};
  v2f a = ap[0];
  v2f b = bp[0];
#pragma unroll 4
  for (int i = 0; i < FCD / 4 - 1; i++) {
    v2f an = ap[(i + 1) * 2];     // prefetch next fragments (stride 4 floats)
    v2f bn = bp[(i + 1) * 2];
    acc = __builtin_amdgcn_wmma_f32_16x16x4_f32(false, a, false, b,
                                                (short)0, acc, false, false);
    a = an;
    b = bn;
  }
  acc = __builtin_amdgcn_wmma_f32_16x16x4_f32(false, a, false, b,
                                              (short)0, acc, false, false);

  int h = h0 + lr;
  float bias = fcb[h];
  float lsum = 0.f, lsq = 0.f;
  float* x3 = ws + X3OFF;
#pragma unroll
  for (int r = 0; r < 8; r++) {
    int row = n0 + r + 8 * hi;
    float val = fmaxf(acc[r] + bias, 0.f);
    if (row < NN) {
      x3[row * HH + h] = val;
      lsum += val;
      lsq  += val * val;
    }
  }
  lsum += __shfl_xor(lsum, 16, 32);
  lsq  += __shfl_xor(lsq, 16, 32);
  if (hi == 0) {
    atomicAdd(&ws[S3SUM + h], lsum);
    atomicAdd(&ws[S3SQ + h], lsq);
  }
}

// ------------------------------------------------------------------
// K6: finalize BN3
// ------------------------------------------------------------------
__global__ void k_fin3(const float* __restrict__ g, const float* __restrict__ b,
                       float* __restrict__ ws) {
  int c = threadIdx.x;
  if (c < HH) {
    float m   = ws[S3SUM + c] / CNT3;
    float var = ws[S3SQ + c] / CNT3 - m * m;
    float s   = g[c] * rsqrtf(var + BN_EPS);
    ws[S3S + c] = s;
    ws[S3T + c] = b[c] - m * s;
  }
}

// ------------------------------------------------------------------
// K6b: x = bn3(x3raw) -> d_out second region
// ------------------------------------------------------------------
__global__ __launch_bounds__(256)
void k_xout(float* __restrict__ out_x, const float* __restrict__ ws) {
  int idx = blockIdx.x * 256 + threadIdx.x;
  if (idx >= NN * HH) return;
  int h = idx & (HH - 1);
  out_x[idx] = fmaf(ws[S3S + h], ws[X3OFF + idx], ws[S3T + h]);
}

// ------------------------------------------------------------------
// K7: a = x @ W2a^T ; rb = x @ W2b^T + fc2_b
// ------------------------------------------------------------------
__global__ __launch_bounds__(256)
void k_ar(const float* __restrict__ w2, const float* __restrict__ b2,
          const float* __restrict__ xfin, float* __restrict__ ws) {
  __shared__ float wl[HH * 2 * HH];  // 64*128 = 8192 floats
  int t = threadIdx.x;
  for (int i = t; i < HH * 2 * HH; i += 256) wl[i] = w2[i];
  __syncthreads();
  int idx = blockIdx.x * 256 + t;   // exactly 96000 threads
  int i = idx & (HH - 1);
  int n = idx >> 6;
  const float* xr = xfin + n * HH;
  float sa = 0.f, sr = 0.f;
#pragma unroll
  for (int j = 0; j < HH; j++) {
    float xv = xr[j];
    sa = fmaf(wl[i * 128 + j],      xv, sa);
    sr = fmaf(wl[i * 128 + HH + j], xv, sr);
  }
  ws[AAOFF + idx] = sa;
  ws[RBOFF + idx] = sr + b2[i];
}

// ------------------------------------------------------------------
// K8: pairwise support tile (64 i x 32 j per block) + row sums
// ------------------------------------------------------------------
__global__ __launch_bounds__(256)
void k_pair(const float* __restrict__ fc1w, const float* __restrict__ fc1b,
            float* __restrict__ ws) {
  __shared__ float As[64 * 65];   // As[h*65 + il]
  __shared__ float Rs[64 * 33];   // Rs[h*33 + jl]
  __shared__ float wsh[HH];
  __shared__ float rpart[64];
  int t = threadIdx.x;
  int i0 = blockIdx.x * 64;
  int j0 = blockIdx.y * 32;
  if (t < HH) { wsh[t] = fc1w[t]; rpart[t] = 0.f; }

  const float* aa = ws + AAOFF;
  const float* rb = ws + RBOFF;
  for (int e = t; e < 64 * 64; e += 256) {
    int il = e >> 6, h = e & 63;
    int gi = i0 + il;
    As[h * 65 + il] = (gi < NN) ? aa[gi * HH + h] : 0.f;
  }
  for (int e = t; e < 32 * 64; e += 256) {
    int jl = e >> 6, h = e & 63;
    int gj = j0 + jl;
    Rs[h * 33 + jl] = (gj < NN) ? rb[gj * HH + h] : -1e30f;
  }
  __syncthreads();

  int tx = t & 15, ty = t >> 4;
  float acc[4][2] = {};
  for (int h = 0; h < HH; h++) {
    float wv = wsh[h];
    float av0 = As[h * 65 + ty];
    float av1 = As[h * 65 + 16 + ty];
    float av2 = As[h * 65 + 32 + ty];
    float av3 = As[h * 65 + 48 + ty];
    float rv0 = Rs[h * 33 + tx];
    float rv1 = Rs[h * 33 + 16 + tx];
    acc[0][0] = fmaf(wv, fmaxf(av0 + rv0, 0.f), acc[0][0]);
    acc[0][1] = fmaf(wv, fmaxf(av0 + rv1, 0.f), acc[0][1]);
    acc[1][0] = fmaf(wv, fmaxf(av1 + rv0, 0.f), acc[1][0]);
    acc[1][1] = fmaf(wv, fmaxf(av1 + rv1, 0.f), acc[1][1]);
    acc[2][0] = fmaf(wv, fmaxf(av2 + rv0, 0.f), acc[2][0]);
    acc[2][1] = fmaf(wv, fmaxf(av2 + rv1, 0.f), acc[2][1]);
    acc[3][0] = fmaf(wv, fmaxf(av3 + rv0, 0.f), acc[3][0]);
    acc[3][1] = fmaf(wv, fmaxf(av3 + rv1, 0.f), acc[3][1]);
  }

  float fb = fc1b[0];
  float* sup = ws + SUPOFF;
  float rs[4] = {0.f, 0.f, 0.f, 0.f};
#pragma unroll
  for (int u = 0; u < 4; u++) {
    int gi = i0 + u * 16 + ty;
    if (gi < NN) {
#pragma unroll
      for (int v = 0; v < 2; v++) {
        int gj = j0 + v * 16 + tx;
        if (gj < NN) {
          float val = acc[u][v] + fb;
          sup[gi * NN + gj] = val;
          rs[u] += val;
        }
      }
    }
  }
#pragma unroll
  for (int u = 0; u < 4; u++) atomicAdd(&rpart[u * 16 + ty], rs[u]);
  __syncthreads();
  if (t < 64) {
    int gi = i0 + t;
    if (gi < NN) atomicAdd(&ws[ROWSUM + gi], rpart[t]);
  }
}

// ------------------------------------------------------------------
// K9: scale by 1/rowsum and broadcast 16x to d_out (float4)
// ------------------------------------------------------------------
__global__ __launch_bounds__(256)
void k_bcast(float* __restrict__ out, const float* __restrict__ ws) {
  int idx = blockIdx.x * 256 + threadIdx.x;   // NN*375 = 562,500
  if (idx >= NN * 375) return;
  int i  = idx / 375;
  int jq = idx % 375;
  const float4* supr = (const float4*)(ws + SUPOFF + i * NN);
  float4 v = supr[jq];
  float rinv = 1.0f / ws[ROWSUM + i];
  v.x *= rinv; v.y *= rinv; v.z *= rinv; v.w *= rinv;
  float4* o = (float4*)out;
  int base = i * 375 + jq;
#pragma unroll
  for (int b = 0; b < BB; b++)
    o[b * 562500 + base] = v;
}

// ------------------------------------------------------------------
extern "C" void kernel_launch(void* const* d_in, const int* in_sizes, int n_in,
                              void* d_out, int out_size, void* d_ws, size_t ws_size,
                              hipStream_t stream) {
  const float* node_fea = (const float*)d_in[0];
  const float* conv1_w  = (const float*)d_in[1];
  const float* conv1_b  = (const float*)d_in[2];
  const float* conv2_w  = (const float*)d_in[3];
  const float* conv2_b  = (const float*)d_in[4];
  const float* bn1_g    = (const float*)d_in[5];
  const float* bn1_b    = (const float*)d_in[6];
  const float* bn2_g    = (const float*)d_in[7];
  const float* bn2_b    = (const float*)d_in[8];
  const float* fc_w     = (const float*)d_in[9];
  const float* fc_b     = (const float*)d_in[10];
  const float* bn3_g    = (const float*)d_in[11];
  const float* bn3_b    = (const float*)d_in[12];
  const float* fc2_w    = (const float*)d_in[13];
  const float* fc2_b    = (const float*)d_in[14];
  const float* fc1_w    = (const float*)d_in[15];
  const float* fc1_b    = (const float*)d_in[16];

  float* out = (float*)d_out;
  float* out_x = out + (size_t)BB * NN * NN;   // second output region
  float* ws = (float*)d_ws;

  // zero stats + rowsum accumulators (harness does not re-zero ws)
  hipMemsetAsync(ws, 0, STATS_FLOATS * sizeof(float), stream);

  k_conv1<<<(NN * L1 + 255) / 256, 256, 0, stream>>>(node_fea, conv1_w, conv1_b, ws);
  k_fin1<<<1, 32, 0, stream>>>(bn1_g, bn1_b, ws);
  k_conv2<<<(NN * L2 + 255) / 256, 256, 0, stream>>>(conv2_w, conv2_b, ws);
  k_fin2<<<1, 32, 0, stream>>>(bn2_g, bn2_b, ws);
  k_bn2<<<(NN * FCD + 255) / 256, 256, 0, stream>>>(ws);
  k_fc_wmma<<<dim3((NN + 15) / 16, HH / 16), 32, 0, stream>>>(fc_w, fc_b, ws);
  k_fin3<<<1, 64, 0, stream>>>(bn3_g, bn3_b, ws);
  k_xout<<<(NN * HH + 255) / 256, 256, 0, stream>>>(out_x, ws);
  k_ar<<<(NN * HH) / 256, 256, 0, stream>>>(fc2_w, fc2_b, out_x, ws);
  k_pair<<<dim3((NN + 63) / 64, (NN + 31) / 32), 256, 0, stream>>>(fc1_w, fc1_b, ws);
  k_bcast<<<(NN * 375 + 255) / 256, 256, 0, stream>>>(out, ws);
}